// DeepFinModel_16243566313769
// MI455X (gfx1250) — compile-verified
//
#include <hip/hip_runtime.h>

typedef __attribute__((ext_vector_type(2))) float v2f;
typedef __attribute__((ext_vector_type(8))) float v8f;

#define NROWS  (1024 * 200)   // B*T
#define DDIM   32
#define NFEAT  12
#define NCAT   4
#define K1DIM  (NFEAT * DDIM) // 384
#define H1     64
#define H2     32
#define BN_EPS 1e-5f

__device__ __forceinline__ v8f wmma4(v2f a, v2f b, v8f c) {
    // D = A(16x4,f32) * B(4x16,f32) + C(16x16,f32)
    return __builtin_amdgcn_wmma_f32_16x16x4_f32(false, a, false, b,
                                                 (short)0, c, false, false);
}

// ---------------------------------------------------------------------------
// K1: fused embedding gather + FM interaction/linear + GEMM1 (384 -> 64)
// block = 256 threads = 8 waves; each wave owns 16 rows.
// ---------------------------------------------------------------------------
__global__ void __launch_bounds__(256) k1_embed_gemm1(
    const float* __restrict__ x,
    const float* __restrict__ emb0, const float* __restrict__ lin0,
    const float* __restrict__ emb1, const float* __restrict__ lin1,
    const float* __restrict__ emb2, const float* __restrict__ lin2,
    const float* __restrict__ emb3, const float* __restrict__ lin3,
    const float* __restrict__ cont_w, const float* __restrict__ cont_b,
    const float* __restrict__ clin_w, const float* __restrict__ clin_b,
    const float* __restrict__ fin_bias,
    const float* __restrict__ w1, const float* __restrict__ b1,
    float* __restrict__ out_base, float* __restrict__ h1out)
{
    __shared__ float xs[8][16 * NFEAT];   // per-wave x tiles
    __shared__ float es[8][16 * DDIM];    // per-wave 16x32 embedding chunk

    const int tid = threadIdx.x, wave = tid >> 5, lane = tid & 31;
    const int rowBase = blockIdx.x * 128 + wave * 16;

    const float* const embT[NCAT] = {emb0, emb1, emb2, emb3};
    const float* const linT[NCAT] = {lin0, lin1, lin2, lin3};

    // stage x tile: 16 rows * 12 feats = 192 = 32 lanes * 6
#pragma unroll
    for (int j = 0; j < 6; ++j) {
        int idx = lane * 6 + j;
        xs[wave][idx] = x[rowBase * NFEAT + idx];
    }
    __syncthreads();

    float s[16], ss[16], lacc[16];
    const float fb = fin_bias[lane];      // lane == embedding dim
#pragma unroll
    for (int r = 0; r < 16; ++r) { s[r] = 0.f; ss[r] = 0.f; lacc[r] = fb; }

    v8f acc[4];
    const v8f vzero = {0.f, 0.f, 0.f, 0.f, 0.f, 0.f, 0.f, 0.f};
#pragma unroll
    for (int t = 0; t < 4; ++t) acc[t] = vzero;

    const int half = lane >> 4, m = lane & 15;

#pragma unroll
    for (int f = 0; f < NFEAT; ++f) {
        float cw = 0.f, cb = 0.f, lw = 0.f, lb = 0.f;
        if (f >= NCAT) {
            int fc = f - NCAT;
            cw = cont_w[fc * DDIM + lane]; cb = cont_b[fc * DDIM + lane];
            lw = clin_w[fc * DDIM + lane]; lb = clin_b[fc * DDIM + lane];
        }
        // build 16x32 chunk (lane = dim), fold FM + linear accumulators
#pragma unroll
        for (int r = 0; r < 16; ++r) {
            float xv = xs[wave][r * NFEAT + f];
            float e, lv;
            if (f < NCAT) {
                int idx = (int)xv;
                e  = embT[f][idx * DDIM + lane];
                lv = linT[f][idx * DDIM + lane];
            } else {
                e  = fmaf(xv, cw, cb);
                lv = fmaf(xv, lw, lb);
            }
            s[r] += e; ss[r] = fmaf(e, e, ss[r]); lacc[r] += lv;
            es[wave][r * DDIM + lane] = e;
        }
        __syncthreads();
        // GEMM over this feature's K=32 chunk: 8 slices of K=4, 4 col tiles
#pragma unroll
        for (int kk = 0; kk < 8; ++kk) {
            int k0 = kk * 4 + 2 * half;
            v2f a = *(const v2f*)&es[wave][m * DDIM + k0];
#pragma unroll
            for (int t = 0; t < 4; ++t) {
                int col = t * 16 + m;
                v2f bf = *(const v2f*)&w1[col * K1DIM + f * DDIM + k0];
                acc[t] = wmma4(a, bf, acc[t]);
            }
        }
        __syncthreads();
    }

    // linear + interaction -> d_out (lane = dim)
#pragma unroll
    for (int r = 0; r < 16; ++r) {
        float v = lacc[r] + 0.5f * (s[r] * s[r] - ss[r]);
        out_base[(size_t)(rowBase + r) * DDIM + lane] = v;
    }
    // pre-BN h1 = acc + b1
#pragma unroll
    for (int t = 0; t < 4; ++t) {
        int col = t * 16 + m;
        float bb = b1[col];
#pragma unroll
        for (int v = 0; v < 8; ++v) {
            int row = rowBase + v + 8 * half;
            h1out[(size_t)row * H1 + col] = acc[t][v] + bb;
        }
    }
}

// ---------------------------------------------------------------------------
// Per-channel sum / sum-of-squares reduction for batchnorm (ncols = 64 or 32)
// stats[0..ncols) = sum, stats[ncols..2*ncols) = sumsq  (pre-zeroed)
// ---------------------------------------------------------------------------
__global__ void __launch_bounds__(256) k_stats(
    const float* __restrict__ h, int ncols, int nrows, float* __restrict__ stats)
{
    __shared__ float ls[256], lq[256];
    const int tid = threadIdx.x;
    const int c = tid & (ncols - 1);
    const int g = tid / ncols;
    const int groups = 256 / ncols;
    float sum = 0.f, sq = 0.f;
    for (int row = blockIdx.x * groups + g; row < nrows; row += gridDim.x * groups) {
        float v = h[(size_t)row * ncols + c];
        sum += v; sq = fmaf(v, v, sq);
    }
    ls[tid] = sum; lq[tid] = sq;
    __syncthreads();
    if (tid < ncols) {
        float ts = 0.f, tq = 0.f;
        for (int gg = 0; gg < groups; ++gg) {
            ts += ls[gg * ncols + tid];
            tq += lq[gg * ncols + tid];
        }
        atomicAdd(&stats[tid], ts);
        atomicAdd(&stats[ncols + tid], tq);
    }
}

// ---------------------------------------------------------------------------
// K3: BN1 + relu while staging to LDS, then GEMM2 (64 -> 32)
// ---------------------------------------------------------------------------
__global__ void __launch_bounds__(256) k3_bn1_gemm2(
    const float* __restrict__ h1, const float* __restrict__ stats1,
    const float* __restrict__ g1, const float* __restrict__ beta1,
    const float* __restrict__ w2, const float* __restrict__ b2,
    float* __restrict__ h2out)
{
    __shared__ float hs[8][16 * H1];
    const int tid = threadIdx.x, wave = tid >> 5, lane = tid & 31;
    const int rowBase = blockIdx.x * 128 + wave * 16;
    const float inv_n = 1.0f / (float)NROWS;

    // lane handles channels lane and lane+32
    float sc0, sh0, sc1, sh1;
    {
        int c0 = lane, c1 = lane + 32;
        float mu0  = stats1[c0] * inv_n;
        float var0 = fmaf(-mu0, mu0, stats1[H1 + c0] * inv_n);
        float is0  = rsqrtf(var0 + BN_EPS);
        sc0 = g1[c0] * is0; sh0 = fmaf(-mu0, sc0, beta1[c0]);
        float mu1  = stats1[c1] * inv_n;
        float var1 = fmaf(-mu1, mu1, stats1[H1 + c1] * inv_n);
        float is1  = rsqrtf(var1 + BN_EPS);
        sc1 = g1[c1] * is1; sh1 = fmaf(-mu1, sc1, beta1[c1]);
    }
#pragma unroll
    for (int r = 0; r < 16; ++r) {
        float v0 = h1[(size_t)(rowBase + r) * H1 + lane];
        float v1 = h1[(size_t)(rowBase + r) * H1 + lane + 32];
        v0 = fmaf(v0, sc0, sh0); v1 = fmaf(v1, sc1, sh1);
        hs[wave][r * H1 + lane]      = v0 > 0.f ? v0 : 0.f;
        hs[wave][r * H1 + lane + 32] = v1 > 0.f ? v1 : 0.f;
    }
    __syncthreads();

    const int half = lane >> 4, m = lane & 15;
    v8f acc[2];
    const v8f vzero = {0.f, 0.f, 0.f, 0.f, 0.f, 0.f, 0.f, 0.f};
    acc[0] = vzero; acc[1] = vzero;
#pragma unroll
    for (int kk = 0; kk < 16; ++kk) {
        int k0 = kk * 4 + 2 * half;
        v2f a = *(const v2f*)&hs[wave][m * H1 + k0];
#pragma unroll
        for (int t = 0; t < 2; ++t) {
            int col = t * 16 + m;
            v2f bf = *(const v2f*)&w2[col * H1 + k0];
            acc[t] = wmma4(a, bf, acc[t]);
        }
    }
#pragma unroll
    for (int t = 0; t < 2; ++t) {
        int col = t * 16 + m;
        float bb = b2[col];
#pragma unroll
        for (int v = 0; v < 8; ++v) {
            int row = rowBase + v + 8 * half;
            h2out[(size_t)row * H2 + col] = acc[t][v] + bb;
        }
    }
}

// ---------------------------------------------------------------------------
// K5: BN2 + relu, GEMM3 (32 -> 32), out += dnn + b_out
// ---------------------------------------------------------------------------
__global__ void __launch_bounds__(256) k5_bn2_gemm3(
    const float* __restrict__ h2, const float* __restrict__ stats2,
    const float* __restrict__ g2, const float* __restrict__ beta2,
    const float* __restrict__ w_out, const float* __restrict__ b_out,
    float* __restrict__ out)
{
    __shared__ float hs[8][16 * H2];
    const int tid = threadIdx.x, wave = tid >> 5, lane = tid & 31;
    const int rowBase = blockIdx.x * 128 + wave * 16;
    const float inv_n = 1.0f / (float)NROWS;

    float mu   = stats2[lane] * inv_n;
    float var  = fmaf(-mu, mu, stats2[H2 + lane] * inv_n);
    float istd = rsqrtf(var + BN_EPS);
    float sc = g2[lane] * istd;
    float sh = fmaf(-mu, sc, beta2[lane]);
#pragma unroll
    for (int r = 0; r < 16; ++r) {
        float v = h2[(size_t)(rowBase + r) * H2 + lane];
        v = fmaf(v, sc, sh);
        hs[wave][r * H2 + lane] = v > 0.f ? v : 0.f;
    }
    __syncthreads();

    const int half = lane >> 4, m = lane & 15;
    v8f acc[2];
    const v8f vzero = {0.f, 0.f, 0.f, 0.f, 0.f, 0.f, 0.f, 0.f};
    acc[0] = vzero; acc[1] = vzero;
#pragma unroll
    for (int kk = 0; kk < 8; ++kk) {
        int k0 = kk * 4 + 2 * half;
        v2f a = *(const v2f*)&hs[wave][m * H2 + k0];
#pragma unroll
        for (int t = 0; t < 2; ++t) {
            int col = t * 16 + m;
            v2f bf = *(const v2f*)&w_out[col * H2 + k0];
            acc[t] = wmma4(a, bf, acc[t]);
        }
    }
#pragma unroll
    for (int t = 0; t < 2; ++t) {
        int col = t * 16 + m;
        float bb = b_out[col];
#pragma unroll
        for (int v = 0; v < 8; ++v) {
            int row = rowBase + v + 8 * half;
            out[(size_t)row * DDIM + col] += acc[t][v] + bb;
        }
    }
}

// ---------------------------------------------------------------------------
extern "C" void kernel_launch(void* const* d_in, const int* in_sizes, int n_in,
                              void* d_out, int out_size, void* d_ws, size_t ws_size,
                              hipStream_t stream)
{
    const float* x       = (const float*)d_in[0];
    const float* emb0    = (const float*)d_in[1];
    const float* lin0    = (const float*)d_in[2];
    const float* emb1    = (const float*)d_in[3];
    const float* lin1    = (const float*)d_in[4];
    const float* emb2    = (const float*)d_in[5];
    const float* lin2    = (const float*)d_in[6];
    const float* emb3    = (const float*)d_in[7];
    const float* lin3    = (const float*)d_in[8];
    const float* cont_w  = (const float*)d_in[9];
    const float* cont_b  = (const float*)d_in[10];
    const float* clin_w  = (const float*)d_in[11];
    const float* clin_b  = (const float*)d_in[12];
    const float* fin_bias= (const float*)d_in[13];
    const float* w1      = (const float*)d_in[14];
    const float* b1      = (const float*)d_in[15];
    const float* g1      = (const float*)d_in[16];
    const float* beta1   = (const float*)d_in[17];
    const float* w2      = (const float*)d_in[18];
    const float* b2      = (const float*)d_in[19];
    const float* g2      = (const float*)d_in[20];
    const float* beta2   = (const float*)d_in[21];
    const float* w_out   = (const float*)d_in[22];
    const float* b_out   = (const float*)d_in[23];
    float* out = (float*)d_out;

    float* ws     = (float*)d_ws;
    float* h1     = ws;                                  // NROWS*64
    float* h2     = ws + (size_t)NROWS * H1;             // NROWS*32
    float* stats1 = ws + (size_t)NROWS * (H1 + H2);      // 128 floats
    float* stats2 = stats1 + 2 * H1;                     // 64 floats

    hipMemsetAsync(stats1, 0, (2 * H1 + 2 * H2) * sizeof(float), stream);

    dim3 blk(256);
    dim3 grd(NROWS / 128);   // 1600 blocks, 16 rows per wave
    k1_embed_gemm1<<<grd, blk, 0, stream>>>(x, emb0, lin0, emb1, lin1, emb2, lin2,
                                            emb3, lin3, cont_w, cont_b, clin_w, clin_b,
                                            fin_bias, w1, b1, out, h1);
    k_stats<<<dim3(512), blk, 0, stream>>>(h1, H1, NROWS, stats1);
    k3_bn1_gemm2<<<grd, blk, 0, stream>>>(h1, stats1, g1, beta1, w2, b2, h2);
    k_stats<<<dim3(512), blk, 0, stream>>>(h2, H2, NROWS, stats2);
    k5_bn2_gemm3<<<grd, blk, 0, stream>>>(h2, stats2, g2, beta2, w_out, b_out, out);
}